// BilinearInterpolation2d_6347961663932
// MI455X (gfx1250) — compile-verified
//
#include <hip/hip_runtime.h>

// Bilinear interpolation over a 4096x4096 fp32 image, 16.7M query points.
// Memory-bound: 256 MB streaming (coords in, values+valid out) + gathers into
// a 64 MB image that fits in MI455X's 192 MB L2. Streaming traffic is marked
// non-temporal (th:NT) so it does not evict the image from L2.

#ifndef IMG_W
#define IMG_W 4096
#define IMG_H 4096
#endif

// Native clang vector type: required by __builtin_nontemporal_{load,store}
typedef float v4f __attribute__((ext_vector_type(4)));

__global__ __launch_bounds__(256) void bilerp2d_kernel(
    const float* __restrict__ img,     // H*W image
    const float* __restrict__ xs,      // N x-coords
    const float* __restrict__ ys,      // N y-coords
    float* __restrict__ out_val,       // N values
    float* __restrict__ out_valid,     // N valid flags (1.0/0.0)
    int n)                             // n >= 4, handled in groups of 4
{
    const float wmax = (float)(IMG_W - 1);
    const float hmax = (float)(IMG_H - 1);

    int i4 = (blockIdx.x * blockDim.x + threadIdx.x) * 4;
    if (i4 >= n) return;
    // Overlap trick: slide a partial last group back so it's always a full
    // float4. Overlapping threads write identical values -> deterministic.
    if (i4 + 4 > n) i4 = n - 4;

    // Non-temporal b128 loads: pure streaming, keep L2 for the image.
    v4f xv = __builtin_nontemporal_load((const v4f*)(xs + i4));
    v4f yv = __builtin_nontemporal_load((const v4f*)(ys + i4));

    v4f vout;
    v4f mout;

#pragma unroll
    for (int j = 0; j < 4; ++j) {
        float xq = xv[j];
        float yq = yv[j];
        float x0 = floorf(xq);
        float y0 = floorf(yq);
        float x1 = x0 + 1.0f;
        float y1 = y0 + 1.0f;

        bool valid = (x0 >= 0.0f) && (x1 <= wmax) &&
                     (y0 >= 0.0f) && (y1 <= hmax);

        // One v_med3_f32 per axis for the clamp, then integer +1/min for the
        // far texel. (For invalid points the gathered indices differ from the
        // reference's clip, but those values are zeroed out anyway.)
        int x0i = (int)__builtin_amdgcn_fmed3f(x0, 0.0f, wmax);
        int y0i = (int)__builtin_amdgcn_fmed3f(y0, 0.0f, hmax);
        int x1i = min(x0i + 1, IMG_W - 1);
        int y1i = min(y0i + 1, IMG_H - 1);

        // Gathers hit L2 (64 MB image resident in 192 MB L2).
        // x1i = x0i+1 -> same cacheline as x0i almost always.
        const float* r0 = img + ((size_t)y0i << 12);
        const float* r1 = img + ((size_t)y1i << 12);
        float v00 = r0[x0i];
        float v10 = r0[x1i];
        float v01 = r1[x0i];
        float v11 = r1[x1i];

        float ax0 = x1 - xq;
        float ax1 = xq - x0;
        float ay0 = y1 - yq;
        float ay1 = yq - y0;

        // Factored blend: 2 row lerps + 1 column lerp (FMA-contracted).
        float w0 = ax0 * v00 + ax1 * v10;
        float w1 = ax0 * v01 + ax1 * v11;
        float v  = ay0 * w0 + ay1 * w1;

        vout[j] = valid ? v : 0.0f;
        mout[j] = valid ? 1.0f : 0.0f;
    }

    // Non-temporal b128 stores: write-once data, bypass-friendly.
    __builtin_nontemporal_store(vout, (v4f*)(out_val + i4));
    __builtin_nontemporal_store(mout, (v4f*)(out_valid + i4));
}

extern "C" void kernel_launch(void* const* d_in, const int* in_sizes, int n_in,
                              void* d_out, int out_size, void* d_ws, size_t ws_size,
                              hipStream_t stream) {
    (void)n_in; (void)d_ws; (void)ws_size; (void)out_size;

    const float* img    = (const float*)d_in[0];  // 4096*4096 image
    const float* coords = (const float*)d_in[1];  // (2, N): x row then y row
    int n = in_sizes[1] / 2;                      // N = 16777216

    const float* xs = coords;
    const float* ys = coords + n;

    float* out_val   = (float*)d_out;        // values: first N floats
    float* out_valid = (float*)d_out + n;    // valid mask: next N floats

    const int threads = 256;                       // 8 wave32 per block
    const int pts_per_block = threads * 4;         // float4 per thread
    const int blocks = (n + pts_per_block - 1) / pts_per_block;

    bilerp2d_kernel<<<blocks, threads, 0, stream>>>(img, xs, ys, out_val, out_valid, n);
}